// PointSIFT_res_module_33071248179392
// MI455X (gfx1250) — compile-verified
//
#include <hip/hip_runtime.h>
#include <hip/hip_bf16.h>

// ---------------------------------------------------------------------------
// PointSIFT res module, MI455X (gfx1250) wave32 + WMMA f32 16x16x4.
// - gather -> layer0 GEMM fused: A-tile built directly in LDS (no X in HBM)
// - layers 1/2: BN scale/shift + ReLU folded into coalesced LDS stage of Y
// - weight panels pre-swizzled into WMMA B-fragment order (1 ds_load_b64 per
//   WMMA B operand), async-staged via GLOBAL_LOAD_ASYNC_TO_LDS_B128
// - batch-norm global stats: ds_add_f32 LDS reduction + 1 global atomic per
//   channel per block, tiny finalize kernel between layers
// X-row layout per K-half (width 68): [feats(64) | dxyz(3) | pad(1)] so the
// gathered 256B feature rows land 16B-aligned in LDS (ds_store_b128).
// ---------------------------------------------------------------------------

#define BB    8
#define NPTS  16384
#define NBITS 14            // log2(NPTS)
#define BN    (BB * NPTS)   // 131072
#define M0C   (BN * 4)      // 524288 rows, layer0 (s: 8 -> 4)
#define M1C   (BN * 2)      // 262144 rows, layer1 (s: 4 -> 2)
#define M2C   (BN)          // 131072 rows, layer2 (s: 2 -> 1)
#define K0P   136           // 2*68 (each half: 64 feats + 3 dxyz + 1 pad)
#define K1P   128           // 2*64
#define KMAX  136
#define EPSV  1e-5f

typedef float v2f __attribute__((ext_vector_type(2)));
typedef float v8f __attribute__((ext_vector_type(8)));

// ------------------- CDNA5 async global->LDS staging -----------------------

__device__ __forceinline__ void async_copy_b128(float* lds_dst, const float* gsrc) {
  // low 32 bits of a generic shared pointer are the LDS byte offset
  unsigned ldsoff = (unsigned)(unsigned long long)lds_dst;
  asm volatile("global_load_async_to_lds_b128 %0, %1, off"
               :: "v"(ldsoff), "v"(gsrc)
               : "memory");
}
__device__ __forceinline__ void wait_async0() {
  asm volatile("s_wait_asynccnt 0x0" ::: "memory");
}

// --------------------------- small utility kernels -------------------------

__global__ void k_zero(float* __restrict__ p, int n) {
  int i = blockIdx.x * blockDim.x + threadIdx.x;
  if (i < n) p[i] = 0.f;
}

__global__ void k_copy(const float* __restrict__ s, float* __restrict__ d, int n) {
  int i = blockIdx.x * blockDim.x + threadIdx.x;
  if (i < n) d[i] = s[i];
}

// Re-pack conv weights w[o][c][p] (OUT=64, Cin, 2) into the WMMA B-fragment
// order used by the GEMM kernels:
//   logical K row kp (matching the X-row layout), channel o
//   -> Wt[(kp>>2)*256 + ((kp>>1)&1)*128 + o*2 + (kp&1)]
// For Cin==67 the logical half layout is [feats(64) | dxyz(3) | pad(1)].
__global__ void k_prepw(const float* __restrict__ w, float* __restrict__ Wt,
                        int Cin, int Kpad) {
  int i = blockIdx.x * blockDim.x + threadIdx.x;
  if (i >= Kpad * 64) return;
  const int kp = i >> 6, o = i & 63;
  const int halfw = (Cin == 67) ? 68 : 64;
  const int p = kp / halfw, q = kp - p * halfw;
  float v = 0.f;
  if (p < 2) {
    int  c     = q;
    bool valid = true;
    if (Cin == 67) {
      if (q < 64)      c = q + 3;       // feats
      else if (q < 67) c = q - 64;      // dxyz
      else             valid = false;   // pad column
    }
    if (valid) v = w[((size_t)o * Cin + c) * 2 + p];
  }
  const int blk = kp >> 2, r = kp & 3, lh = r >> 1, e = r & 1;
  Wt[(size_t)blk * 256 + lh * 128 + o * 2 + e] = v;
}

// stats -> (scale, shift): y_norm = y*scale + shift
__global__ __launch_bounds__(64)
void k_statsf(float* __restrict__ st, const float* __restrict__ gamma,
              const float* __restrict__ beta, float invM) {
  const int t = threadIdx.x;
  const float m  = st[t] * invM;
  const float v  = st[64 + t] * invM - m * m;
  const float sc = gamma[t] * rsqrtf(v + EPSV);
  st[128 + t] = sc;
  st[192 + t] = beta[t] - m * sc;
}

// new_points after stack 1: relu(bn(Y2[:,0,:]))
__global__ void k_np1(const float* __restrict__ Y, const float* __restrict__ st,
                      float* __restrict__ np) {
  int i = blockIdx.x * blockDim.x + threadIdx.x;
  if (i >= BN * 64) return;
  const int ch = i & 63;
  np[i] = fmaxf(Y[i] * st[128 + ch] + st[192 + ch], 0.f);
}

// final: relu(bn_noact(Y2b[:,0,:]) + points)
__global__ void k_outnp(const float* __restrict__ Y, const float* __restrict__ st,
                        const float* __restrict__ pts, float* __restrict__ out) {
  int i = blockIdx.x * blockDim.x + threadIdx.x;
  if (i >= BN * 64) return;
  const int ch = i & 63;
  out[i] = fmaxf(Y[i] * st[128 + ch] + st[192 + ch] + pts[i], 0.f);
}

// ---------------------- shared GEMM core + epilogue ------------------------

__device__ __forceinline__ void wmma_step(const float* __restrict__ ar, int kc,
                                          const float* __restrict__ sWb,
                                          int l15, v8f acc[4]) {
  v2f a; a.x = ar[kc]; a.y = ar[kc + 1];   // contiguous -> single ds b64
#pragma unroll
  for (int nt = 0; nt < 4; ++nt) {
    const v2f bf = *(const v2f*)&sWb[(nt * 16 + l15) * 2];  // swizzled panel
    acc[nt] = __builtin_amdgcn_wmma_f32_16x16x4_f32(
        false, a, false, bf, (short)0, acc[nt], false, false);
  }
}

__device__ __forceinline__ void epilogue(v8f acc[4], float* __restrict__ Y,
                                         float* __restrict__ stats,
                                         float* __restrict__ ssum,
                                         float* __restrict__ ssq,
                                         int row0, int lh, int l15, int tid) {
#pragma unroll
  for (int nt = 0; nt < 4; ++nt) {
    const int n = nt * 16 + l15;
    float s = 0.f, q = 0.f;
#pragma unroll
    for (int i = 0; i < 8; ++i) {           // C/D: VGPR i -> M = i + 8*lh
      const int   m = row0 + lh * 8 + i;
      const float v = acc[nt][i];
      Y[(size_t)m * 64 + n] = v;
      s += v; q += v * v;
    }
    atomicAdd(&ssum[n], s);                 // ds_add_f32
    atomicAdd(&ssq[n], q);
  }
  __syncthreads();
  if (tid < 64) {
    atomicAdd(&stats[tid], ssum[tid]);
    atomicAdd(&stats[64 + tid], ssq[tid]);
  }
}

// ----------------- fused gather + layer-0 GEMM + stats ---------------------
// Block = 128 threads (4 waves) handles 16 points -> 64 A rows of K=136,
// gathered straight into LDS. Weights async-staged to LDS concurrently.

__global__ __launch_bounds__(128)
void k_gemm_g(const float* __restrict__ xyz, const float* __restrict__ feat,
              const int* __restrict__ idx, float* __restrict__ gx,
              const float* __restrict__ Wt, float* __restrict__ Y,
              float* __restrict__ stats, int firstStack) {
  __shared__ alignas(16) float sW[K0P * 64];   // 34816 B, B-fragment order
  __shared__ alignas(16) float sA[64 * K0P];   // 34816 B
  __shared__ float ssum[64], ssq[64];

  const int tid = threadIdx.x;
  const int bn0 = blockIdx.x * 16;

  // kick off async weight staging (ASYNCcnt)
  for (int i = tid * 4; i < K0P * 64; i += 128 * 4)
    async_copy_b128(&sW[i], &Wt[i]);

  // gather: one thread per (point, neighbor) pair; 16*8 = 128 pairs
  {
    const int bnl = tid >> 3, j = tid & 7;
    const int bn  = bn0 + bnl;
    const int b   = bn >> NBITS;
    const int nb  = idx[(size_t)bn * 8 + j];
    const size_t g = (size_t)b * NPTS + nb;
    const int row = bnl * 4 + (j >> 1);
    float* ar = &sA[row * K0P + (j & 1) * 68];          // 16B-aligned
    const float4* f4 = (const float4*)(feat + g * 64);  // 256B-aligned row
#pragma unroll
    for (int c4 = 0; c4 < 16; ++c4)
      *(float4*)&ar[4 * c4] = f4[c4];                   // ds_store_b128
    if (firstStack) {
#pragma unroll
      for (int k = 0; k < 3; ++k) {
        const float d = xyz[g * 3 + k] - xyz[(size_t)bn * 3 + k];
        ar[64 + k] = d;
        gx[((size_t)bn * 8 + j) * 3 + k] = d;           // reuse in stack 2
      }
    } else {
#pragma unroll
      for (int k = 0; k < 3; ++k) ar[64 + k] = gx[((size_t)bn * 8 + j) * 3 + k];
    }
    ar[67] = 0.f;                                       // pad column
  }
  if (tid < 64) { ssum[tid] = 0.f; ssq[tid] = 0.f; }
  wait_async0();
  __syncthreads();

  const int wave = tid >> 5, lane = tid & 31;
  const int lh = lane >> 4, l15 = lane & 15;
  const float* ar = &sA[(wave * 16 + l15) * K0P];

  v8f acc[4] = {};
  for (int kb = 0; kb < K0P; kb += 4)
    wmma_step(ar, kb + 2 * lh, &sW[kb * 64 + lh * 128], l15, acc);

  epilogue(acc, Y, stats, ssum, ssq, blockIdx.x * 64 + wave * 16, lh, l15, tid);
}

// ------------- layers 1/2: BN(prev)+ReLU folded into LDS stage -------------
// Block = 128 threads handles 64 output rows; its input is the contiguous
// 128-row slab Yin[blk*128 .. +128) -> coalesced float4 stage with transform.

__global__ __launch_bounds__(128)
void k_gemm_bn(const float* __restrict__ Yin, const float* __restrict__ stp,
               const float* __restrict__ Wt, float* __restrict__ Yout,
               float* __restrict__ stats) {
  __shared__ alignas(16) float sW[K1P * 64];   // 32768 B, B-fragment order
  __shared__ alignas(16) float sA[128 * 64];   // 32768 B
  __shared__ float ssc[64], ssh[64], ssum[64], ssq[64];

  const int tid = threadIdx.x;

  for (int i = tid * 4; i < K1P * 64; i += 128 * 4)
    async_copy_b128(&sW[i], &Wt[i]);

  if (tid < 64) {
    ssc[tid] = stp[128 + tid]; ssh[tid] = stp[192 + tid];
    ssum[tid] = 0.f; ssq[tid] = 0.f;
  }
  __syncthreads();                         // ssc/ssh visible for transform

  const float4* y4 = (const float4*)(Yin + (size_t)blockIdx.x * 128 * 64);
  for (int e = tid * 4; e < 128 * 64; e += 128 * 4) {
    float4 v = y4[e >> 2];
    const int c = e & 63;
    v.x = fmaxf(v.x * ssc[c + 0] + ssh[c + 0], 0.f);
    v.y = fmaxf(v.y * ssc[c + 1] + ssh[c + 1], 0.f);
    v.z = fmaxf(v.z * ssc[c + 2] + ssh[c + 2], 0.f);
    v.w = fmaxf(v.w * ssc[c + 3] + ssh[c + 3], 0.f);
    *(float4*)&sA[e] = v;
  }
  wait_async0();
  __syncthreads();

  const int wave = tid >> 5, lane = tid & 31;
  const int lh = lane >> 4, l15 = lane & 15;
  const int rloc = wave * 16 + l15;        // local output row 0..63

  v8f acc[4] = {};
  for (int kb = 0; kb < K1P; kb += 4) {
    const int kc   = kb + 2 * lh;
    const int srow = 2 * rloc + (kc >> 6); // input row pair (s-axis conv)
    wmma_step(&sA[srow * 64 - (kc & ~63)], kc, &sW[kb * 64 + lh * 128], l15, acc);
  }

  epilogue(acc, Yout, stats, ssum, ssq, blockIdx.x * 64 + wave * 16, lh, l15, tid);
}

// --------------------------- host orchestration ----------------------------

extern "C" void kernel_launch(void* const* d_in, const int* in_sizes, int n_in,
                              void* d_out, int out_size, void* d_ws, size_t ws_size,
                              hipStream_t stream) {
  (void)in_sizes; (void)n_in; (void)out_size; (void)ws_size;

  const float* xyz = (const float*)d_in[0];
  const float* pts = (const float*)d_in[1];
  const int*   idx = (const int*)d_in[2];
  const float* cw[6] = { (const float*)d_in[3],  (const float*)d_in[6],
                         (const float*)d_in[9],  (const float*)d_in[12],
                         (const float*)d_in[15], (const float*)d_in[18] };
  const float* cg[6] = { (const float*)d_in[4],  (const float*)d_in[7],
                         (const float*)d_in[10], (const float*)d_in[13],
                         (const float*)d_in[16], (const float*)d_in[19] };
  const float* cb[6] = { (const float*)d_in[5],  (const float*)d_in[8],
                         (const float*)d_in[11], (const float*)d_in[14],
                         (const float*)d_in[17], (const float*)d_in[20] };
  const int cin[6]  = { 67, 64, 64, 67, 64, 64 };
  const int kpad[6] = { K0P, K1P, K1P, K0P, K1P, K1P };

  float* ws = (float*)d_ws;
  float* Y0 = ws;                           // M0C*64
  float* Y1 = Y0 + (size_t)M0C * 64;        // M1C*64
  float* Y2 = Y1 + (size_t)M1C * 64;        // M2C*64
  float* GX = Y2 + (size_t)M2C * 64;        // BN*8*3
  float* NP = GX + (size_t)BN * 24;         // BN*64
  float* WT = NP + (size_t)BN * 64;         // 6 * KMAX*64
  float* ST = WT + (size_t)6 * KMAX * 64;   // 6 * 256

  k_zero<<<(6 * 256 + 255) / 256, 256, 0, stream>>>(ST, 6 * 256);
  for (int l = 0; l < 6; ++l) {
    const int n = kpad[l] * 64;
    k_prepw<<<(n + 255) / 256, 256, 0, stream>>>(cw[l], WT + (size_t)l * KMAX * 64,
                                                 cin[l], kpad[l]);
  }

  float* out = (float*)d_out;

  for (int stack = 0; stack < 2; ++stack) {
    float* st0 = ST + (size_t)(stack * 3 + 0) * 256;
    float* st1 = ST + (size_t)(stack * 3 + 1) * 256;
    float* st2 = ST + (size_t)(stack * 3 + 2) * 256;
    const float* w0 = WT + (size_t)(stack * 3 + 0) * KMAX * 64;
    const float* w1 = WT + (size_t)(stack * 3 + 1) * KMAX * 64;
    const float* w2 = WT + (size_t)(stack * 3 + 2) * KMAX * 64;

    k_gemm_g<<<BN / 16, 128, 0, stream>>>(xyz, stack == 0 ? pts : NP, idx, GX,
                                          w0, Y0, st0, stack == 0 ? 1 : 0);
    k_statsf<<<1, 64, 0, stream>>>(st0, cg[stack * 3 + 0], cb[stack * 3 + 0],
                                   1.f / (float)M0C);

    k_gemm_bn<<<M1C / 64, 128, 0, stream>>>(Y0, st0, w1, Y1, st1);
    k_statsf<<<1, 64, 0, stream>>>(st1, cg[stack * 3 + 1], cb[stack * 3 + 1],
                                   1.f / (float)M1C);

    k_gemm_bn<<<M2C / 64, 128, 0, stream>>>(Y1, st1, w2, Y2, st2);
    k_statsf<<<1, 64, 0, stream>>>(st2, cg[stack * 3 + 2], cb[stack * 3 + 2],
                                   1.f / (float)M2C);

    if (stack == 0) {
      k_np1<<<(BN * 64 + 255) / 256, 256, 0, stream>>>(Y2, st2, NP);
    } else {
      k_copy<<<(BN * 3 + 255) / 256, 256, 0, stream>>>(xyz, out, BN * 3);
      k_outnp<<<(BN * 64 + 255) / 256, 256, 0, stream>>>(Y2, st2, pts,
                                                         out + (size_t)BN * 3);
    }
  }
}